// LIL_Loss_30081950941623
// MI455X (gfx1250) — compile-verified
//
#include <hip/hip_runtime.h>
#include <hip/hip_bf16.h>

typedef __attribute__((ext_vector_type(2))) float v2f;
typedef __attribute__((ext_vector_type(8))) float v8f;

#define BQ   1024        // batch (rows)
#define NQ   32768       // queue length (cols)
#define DQ   128         // feature dim
#define INV_T 14.285714285714286f   // 1 / 0.07

// ---------------------------------------------------------------------------
// Kernel 1: logits[BQ x NQ] = (output_q @ queue^T) / T  via V_WMMA_F32_16X16X4_F32
// Each wave computes a 16(M) x 64(N) tile; block = 4 waves => 16 x 256 per block.
// ---------------------------------------------------------------------------
__global__ __launch_bounds__(128) void lil_gemm_wmma(
    const float* __restrict__ A,     // output_q [BQ x DQ]
    const float* __restrict__ Q,     // queue    [NQ x DQ]  (= queue_all + BQ*DQ)
    float* __restrict__ C)           // logits   [BQ x NQ]
{
  const int lane = threadIdx.x & 31;
  const int wave = threadIdx.x >> 5;
  const int m0 = blockIdx.y * 16;
  const int n0 = blockIdx.x * 256 + wave * 64;

  const int r  = lane & 15;   // A: row m-offset; B: col n-offset (same per-lane form)
  const int kh = lane >> 4;   // K-half select (K pairs {0,1} vs {2,3})

  const float* Arow = A + (size_t)(m0 + r) * DQ + 2 * kh;
  const float* Qrow = Q + (size_t)(n0 + r) * DQ + 2 * kh;

  v8f acc0 = {}, acc1 = {}, acc2 = {}, acc3 = {};

#pragma unroll
  for (int t = 0; t < 32; ++t) {
    const int kb = 4 * t;
    v2f a  = *(const v2f*)(Arow + kb);
    v2f b0 = *(const v2f*)(Qrow + kb);
    v2f b1 = *(const v2f*)(Qrow + 16 * DQ + kb);
    v2f b2 = *(const v2f*)(Qrow + 32 * DQ + kb);
    v2f b3 = *(const v2f*)(Qrow + 48 * DQ + kb);
    acc0 = __builtin_amdgcn_wmma_f32_16x16x4_f32(false, a, false, b0, (short)0, acc0, false, false);
    acc1 = __builtin_amdgcn_wmma_f32_16x16x4_f32(false, a, false, b1, (short)0, acc1, false, false);
    acc2 = __builtin_amdgcn_wmma_f32_16x16x4_f32(false, a, false, b2, (short)0, acc2, false, false);
    acc3 = __builtin_amdgcn_wmma_f32_16x16x4_f32(false, a, false, b3, (short)0, acc3, false, false);
  }

  // C/D layout: acc[v] -> (M = m0 + v + 8*(lane>>4), N = n0 + (lane&15))
  const int n  = lane & 15;
  const int mo = 8 * (lane >> 4);
  float* Crow = C + (size_t)(m0 + mo) * NQ + n0 + n;
#pragma unroll
  for (int v = 0; v < 8; ++v) {
    Crow[(size_t)v * NQ +  0] = acc0[v] * INV_T;
    Crow[(size_t)v * NQ + 16] = acc1[v] * INV_T;
    Crow[(size_t)v * NQ + 32] = acc2[v] * INV_T;
    Crow[(size_t)v * NQ + 48] = acc3[v] * INV_T;
  }
}

// ---------------------------------------------------------------------------
// Block reductions (256 threads)
// ---------------------------------------------------------------------------
__device__ __forceinline__ float blockReduceSum256(float v, float* s) {
  const int tid = threadIdx.x;
  s[tid] = v; __syncthreads();
#pragma unroll
  for (int off = 128; off > 0; off >>= 1) {
    if (tid < off) s[tid] += s[tid + off];
    __syncthreads();
  }
  float r = s[0]; __syncthreads();
  return r;
}

__device__ __forceinline__ float blockReduceMax256(float v, float* s) {
  const int tid = threadIdx.x;
  s[tid] = v; __syncthreads();
#pragma unroll
  for (int off = 128; off > 0; off >>= 1) {
    if (tid < off) s[tid] = fmaxf(s[tid], s[tid + off]);
    __syncthreads();
  }
  float r = s[0]; __syncthreads();
  return r;
}

// ---------------------------------------------------------------------------
// Kernel 2: one block per row; 3 sweeps over the 32768-wide row (stays in L2).
//   sweep1: row max
//   sweep2: sum(lfp), count(lfp!=0), sum(exp(l) for l!=0 & !col_m1)
//   sweep3: pos/wrong counts of lfp vs mean  -> revise
// Writes revised flags (as float) to outFlags and per-row stats to R.
// ---------------------------------------------------------------------------
__global__ __launch_bounds__(256) void lil_row_pass(
    const float* __restrict__ L,          // logits [BQ x NQ]
    const int*   __restrict__ flags,      // [BQ]
    const int*   __restrict__ qflagsAll,  // [BQ + NQ]
    const int*   __restrict__ nIter,      // scalar
    float* __restrict__ R,                // [BQ x 4]: sum_lfp_final, nnz_final, sumexp
    float* __restrict__ outFlags)         // d_out + 1
{
  __shared__ float sred[256];
  const int i   = blockIdx.x;
  const int tid = threadIdx.x;
  const float* row = L + (size_t)i * NQ;
  const int*   qf  = qflagsAll + BQ;

  const int  fi    = flags[i];
  const bool rowm1 = (fi == -1);

  // ---- sweep 1: row max of raw logits ----
  float mx = -3.402823466e38f;
  for (int j = tid * 4; j < NQ; j += 1024) {
    float4 x = *(const float4*)(row + j);
    mx = fmaxf(mx, fmaxf(fmaxf(x.x, x.y), fmaxf(x.z, x.w)));
  }
  mx = blockReduceMax256(mx, sred);

  // ---- sweep 2: sum_lfp, nnz, sum_exp ----
  float slfp = 0.f, nnz = 0.f, sexp = 0.f;
  for (int j = tid * 4; j < NQ; j += 1024) {
    float4 x = *(const float4*)(row + j);
    int4   q = *(const int4*)(qf + j);
    float xs[4] = {x.x, x.y, x.z, x.w};
    int   qs[4] = {q.x, q.y, q.z, q.w};
#pragma unroll
    for (int c = 0; c < 4; ++c) {
      float l   = rowm1 ? 0.f : (xs[c] - mx);          // 0 exactly at argmax
      bool  mt  = (!rowm1) && (fi == qs[c]);
      float lfp = mt ? l : 0.f;
      slfp += lfp;
      nnz  += (lfp != 0.f) ? 1.f : 0.f;
      float le = (rowm1 || qs[c] == -1) ? 0.f : l;     // col_m1 zeroing
      if (le != 0.f) sexp += __expf(le);               // exp(l)*(exp(l)!=1)
    }
  }
  slfp = blockReduceSum256(slfp, sred);
  nnz  = blockReduceSum256(nnz,  sred);
  sexp = blockReduceSum256(sexp, sred);
  const float mean = slfp * (1.0f / (float)NQ);

  // ---- sweep 3: pos/wrong counts vs mean ----
  float pos = 0.f, wrong = 0.f;
  for (int j = tid * 4; j < NQ; j += 1024) {
    float4 x = *(const float4*)(row + j);
    int4   q = *(const int4*)(qf + j);
    float xs[4] = {x.x, x.y, x.z, x.w};
    int   qs[4] = {q.x, q.y, q.z, q.w};
#pragma unroll
    for (int c = 0; c < 4; ++c) {
      float l   = rowm1 ? 0.f : (xs[c] - mx);
      bool  mt  = (!rowm1) && (fi == qs[c]);
      float lfp = mt ? l : 0.f;
      if (lfp >= mean) pos += 1.f; else wrong += 1.f;
    }
  }
  pos   = blockReduceSum256(pos,   sred);
  wrong = blockReduceSum256(wrong, sred);

  if (tid == 0) {
    bool revise = (nIter[0] >= 0) && ((wrong / (pos + 1e-6f)) > 10.0f);
    R[(size_t)i * 4 + 0] = revise ? 0.f : slfp;
    R[(size_t)i * 4 + 1] = revise ? 0.f : nnz;
    R[(size_t)i * 4 + 2] = sexp;
    R[(size_t)i * 4 + 3] = 0.f;
    outFlags[i] = (float)(revise ? 0 : fi);
  }
}

// ---------------------------------------------------------------------------
// Kernel 3: single block of 1024 threads; global nnz reduction + loss.
// ---------------------------------------------------------------------------
__global__ __launch_bounds__(1024) void lil_finalize(
    const float* __restrict__ R, float* __restrict__ out)
{
  __shared__ float s[1024];
  const int i = threadIdx.x;
  const float slfp = R[(size_t)i * 4 + 0];
  const float nnz  = R[(size_t)i * 4 + 1];
  const float sexp = R[(size_t)i * 4 + 2];

  // reduce total nonzero count
  s[i] = nnz; __syncthreads();
#pragma unroll
  for (int off = 512; off > 0; off >>= 1) {
    if (i < off) s[i] += s[i + off];
    __syncthreads();
  }
  const float num_nonzero = s[0]; __syncthreads();

  const float row_lp = slfp - (float)NQ * logf(sexp + 1e-12f);
  const bool  cond   = (num_nonzero > 0.f) && (sexp > 0.f) && (row_lp < 0.f);
  float contrib = cond ? (row_lp / num_nonzero) : 0.f;

  s[i] = contrib; __syncthreads();
#pragma unroll
  for (int off = 512; off > 0; off >>= 1) {
    if (i < off) s[i] += s[i + off];
    __syncthreads();
  }
  if (i == 0) out[0] = -(s[0] / (float)BQ);   // TEMPERATURE/BASE_TEMPERATURE == 1
}

// ---------------------------------------------------------------------------
extern "C" void kernel_launch(void* const* d_in, const int* in_sizes, int n_in,
                              void* d_out, int out_size, void* d_ws, size_t ws_size,
                              hipStream_t stream) {
  const float* output_q   = (const float*)d_in[0];   // [1024 x 128]
  const float* queue_all  = (const float*)d_in[1];   // [33792 x 128]
  const int*   flags      = (const int*)  d_in[2];   // [1024]
  const int*   qflags_all = (const int*)  d_in[3];   // [33792]
  const int*   n_iter     = (const int*)  d_in[4];   // scalar

  float* out    = (float*)d_out;                  // [0]=loss, [1..1024]=flags
  float* logits = (float*)d_ws;                   // 1024*32768 f32 = 128 MB
  float* R      = logits + (size_t)BQ * NQ;       // 1024*4 f32 scratch

  dim3 g1(NQ / 256, BQ / 16);                     // 128 x 64 blocks
  lil_gemm_wmma<<<g1, 128, 0, stream>>>(output_q, queue_all + (size_t)BQ * DQ, logits);

  lil_row_pass<<<BQ, 256, 0, stream>>>(logits, flags, qflags_all, n_iter, R, out + 1);

  lil_finalize<<<1, 1024, 0, stream>>>(R, out);
}